// GCNClassifier_24223615550098
// MI455X (gfx1250) — compile-verified
//
#include <hip/hip_runtime.h>
#include <hip/hip_bf16.h>
#include <math.h>
#include <stdint.h>

typedef float v2f __attribute__((ext_vector_type(2)));
typedef float v8f __attribute__((ext_vector_type(8)));

__device__ __forceinline__ void atomic_add_f32(float* p, float v) {
  // relaxed agent-scope fp32 add -> global_atomic_add_f32 (no return)
  __hip_atomic_fetch_add(p, v, __ATOMIC_RELAXED, __HIP_MEMORY_SCOPE_AGENT);
}

__global__ void k_zero(float* __restrict__ p, long long n) {
  long long i = (long long)blockIdx.x * blockDim.x + threadIdx.x;
  long long stride = (long long)gridDim.x * blockDim.x;
  for (; i < n; i += stride) p[i] = 0.0f;
}

__global__ void k_degree(const int* __restrict__ dst, float* __restrict__ deg, int E) {
  int i = blockIdx.x * blockDim.x + threadIdx.x;
  int stride = gridDim.x * blockDim.x;
  for (; i < E; i += stride) atomic_add_f32(&deg[dst[i]], 1.0f);
}

__global__ void k_rsqrt_deg(float* __restrict__ d, int n) {
  int i = blockIdx.x * blockDim.x + threadIdx.x;
  if (i < n) d[i] = rsqrtf(d[i] + 1.0f);
}

// Pack W (K x F row-major) into WtPad (Fpad x Kpad, transposed, zero-padded).
__global__ void k_pack_wt(const float* __restrict__ W, float* __restrict__ WtPad,
                          int K, int F, int Kpad, int Fpad) {
  int i = blockIdx.x * blockDim.x + threadIdx.x;
  int tot = Kpad * Fpad;
  if (i >= tot) return;
  int f = i / Kpad, k = i - f * Kpad;
  WtPad[i] = (k < K && f < F) ? W[k * F + f] : 0.0f;
}

// H[row,f] = sum_k X[row,k] * W[k,f]
// X: nRows x K (row-major); WtPad: Fpad x Kpad transposed+padded (from k_pack_wt).
// 128 threads = 4 waves; each wave owns one 16-row tile (64 rows / block).
// LDS staging uses global_load_async_to_lds_b128 (ASYNCcnt) on the contiguous
// fast paths; inner loop is branch-free ds_load_b64 pairs + WMMA.
__global__ __launch_bounds__(128) void k_gemm_wmma(
    const float* __restrict__ X, const float* __restrict__ WtPad,
    float* __restrict__ H, int nRows, int K, int F) {
  __shared__ __align__(16) float sX[64 * 80];    // [localRow][Kpad]   (20 KB)
  __shared__ __align__(16) float sWt[80 * 80];   // [f][Kpad] = W^T    (25.6 KB)
  const int Kpad = (K + 3) & ~3;
  const int Fpad = (F + 15) & ~15;
  const int blockRow0 = blockIdx.x * 64;

  // ---- stage transposed/padded W: flat async copy of Fpad*Kpad floats
  {
    const int totB = Fpad * Kpad * 4;                  // bytes, multiple of 16
    const unsigned lds0 = (unsigned)(uintptr_t)(void*)sWt;
    for (int ob = threadIdx.x * 16; ob < totB; ob += blockDim.x * 16) {
      unsigned ldsAddr = lds0 + (unsigned)ob;
      unsigned gOff = (unsigned)ob;
      asm volatile("global_load_async_to_lds_b128 %0, %1, %2"
                   :: "v"(ldsAddr), "v"(gOff), "s"(WtPad) : "memory");
    }
  }
  // ---- stage X row tile
  const bool fastX = ((K & 3) == 0) && (blockRow0 + 64 <= nRows);  // block-uniform
  if (fastX) {
    const int totB = 64 * K * 4;                       // contiguous rows, mult of 16
    const unsigned lds0 = (unsigned)(uintptr_t)(void*)sX;
    const float* Xblk = X + (size_t)blockRow0 * K;
    for (int ob = threadIdx.x * 16; ob < totB; ob += blockDim.x * 16) {
      unsigned ldsAddr = lds0 + (unsigned)ob;
      unsigned gOff = (unsigned)ob;
      asm volatile("global_load_async_to_lds_b128 %0, %1, %2"
                   :: "v"(ldsAddr), "v"(gOff), "s"(Xblk) : "memory");
    }
  } else {
    for (int idx = threadIdx.x; idx < 64 * Kpad; idx += blockDim.x) {
      int r = idx / Kpad, k = idx - r * Kpad;
      int row = blockRow0 + r;
      sX[idx] = (row < nRows && k < K) ? X[(size_t)row * K + k] : 0.0f;
    }
  }
  asm volatile("s_wait_asynccnt 0" ::: "memory");
  __syncthreads();

  const int lane = threadIdx.x & 31;
  const int wid  = threadIdx.x >> 5;
  const int row0 = blockRow0 + wid * 16;
  if (row0 >= nRows) return;             // wave-uniform exit (EXEC all-1 for WMMA)

  const int m  = lane & 15;              // M index (A) / N index (B)
  const int kh = (lane >> 4) << 1;       // K sub-offset: 0 (lanes 0-15) or 2 (16-31)
  const float* aRow = &sX[(wid * 16 + m) * Kpad + kh];   // 8B-aligned pairs

  for (int ct = 0; ct < Fpad; ct += 16) {
    const float* bRow = &sWt[(ct + m) * Kpad + kh];
    v8f acc = {0.f, 0.f, 0.f, 0.f, 0.f, 0.f, 0.f, 0.f};
    for (int k = 0; k < Kpad; k += 4) {
      v2f a = *reinterpret_cast<const v2f*>(aRow + k);   // ds_load_b64
      v2f b = *reinterpret_cast<const v2f*>(bRow + k);   // ds_load_b64
      acc = __builtin_amdgcn_wmma_f32_16x16x4_f32(
          /*neg_a=*/false, a, /*neg_b=*/false, b,
          /*c_mod=*/(short)0, acc, /*reuse_a=*/false, /*reuse_b=*/false);
    }
    // D layout: VGPR r -> row (lane<16 ? r : r+8), col = ct + (lane&15)
    const int outRowBase = row0 + ((lane < 16) ? 0 : 8);
    const int col = ct + m;
    if (col < F) {
      #pragma unroll
      for (int r = 0; r < 8; ++r) {
        int orow = outRowBase + r;
        if (orow < nRows) H[(size_t)orow * F + col] = acc[r];
      }
    }
  }
}

// agg[dst] += h[src] * dinv[src]*dinv[dst], 4 features per thread
__global__ void k_aggregate(const float* __restrict__ h, const int* __restrict__ src,
                            const int* __restrict__ dst, const float* __restrict__ dinv,
                            float* __restrict__ agg, int E, int F, int chunks) {
  long long i = (long long)blockIdx.x * blockDim.x + threadIdx.x;
  long long total  = (long long)E * chunks;
  long long stride = (long long)gridDim.x * blockDim.x;
  for (; i < total; i += stride) {
    int e = (int)(i / chunks);
    int c = (int)(i - (long long)e * chunks);
    int s = src[e], d = dst[e];
    float norm = dinv[s] * dinv[d];
    int f0 = c * 4;
    const float* hp = h   + (size_t)s * F + f0;
    float*       ap = agg + (size_t)d * F + f0;
    int rem = F - f0;
    int cnt = rem < 4 ? rem : 4;
    for (int j = 0; j < cnt; ++j) atomic_add_f32(ap + j, hp[j] * norm);
  }
}

// agg = agg + h*dinv^2 + bias, optional leaky_relu(0.01), in place on agg
__global__ void k_selfloop_bias_act(const float* __restrict__ h, const float* __restrict__ dinv,
                                    const float* __restrict__ bias, float* __restrict__ agg,
                                    int nNodes, int F, int doRelu) {
  long long i = (long long)blockIdx.x * blockDim.x + threadIdx.x;
  long long total  = (long long)nNodes * F;
  long long stride = (long long)gridDim.x * blockDim.x;
  for (; i < total; i += stride) {
    int n = (int)(i / F);
    int f = (int)(i - (long long)n * F);
    float di = dinv[n];
    float v = agg[i] + h[i] * (di * di) + bias[f];
    if (doRelu) v = (v > 0.0f) ? v : 0.01f * v;
    agg[i] = v;
  }
}

// out = log_softmax(logits + bias) over 10 classes, one node per thread
__global__ void k_logsoftmax10(const float* __restrict__ logits, const float* __restrict__ bias,
                               float* __restrict__ out, int n) {
  int i = blockIdx.x * blockDim.x + threadIdx.x;
  if (i >= n) return;
  const float* L = logits + (size_t)i * 10;
  float v[10];
  #pragma unroll
  for (int c = 0; c < 10; ++c) v[c] = L[c] + bias[c];
  float mx = v[0];
  #pragma unroll
  for (int c = 1; c < 10; ++c) mx = fmaxf(mx, v[c]);
  float s = 0.0f;
  #pragma unroll
  for (int c = 0; c < 10; ++c) s += expf(v[c] - mx);
  float ls = logf(s);
  float* O = out + (size_t)i * 10;
  #pragma unroll
  for (int c = 0; c < 10; ++c) O[c] = v[c] - mx - ls;
}

extern "C" void kernel_launch(void* const* d_in, const int* in_sizes, int n_in,
                              void* d_out, int out_size, void* d_ws, size_t ws_size,
                              hipStream_t stream) {
  const float* x  = (const float*)d_in[0];
  const int*   ei = (const int*)d_in[1];
  const float* W1 = (const float*)d_in[2];
  const float* b1 = (const float*)d_in[3];
  const float* W2 = (const float*)d_in[4];
  const float* b2 = (const float*)d_in[5];
  const float* W3 = (const float*)d_in[6];
  const float* b3 = (const float*)d_in[7];
  const float* Wl = (const float*)d_in[8];
  const float* bl = (const float*)d_in[9];

  const int N = in_sizes[0] / 64;   // 100000
  const int E = in_sizes[1] / 2;    // 1000000
  const int* src = ei;
  const int* dst = ei + E;

  // workspace: dinv[N] | wtpad[6656] | bufA[N*80] | bufB[N*80]  (~64.5 MB)
  float* ws    = (float*)d_ws;
  float* dinv  = ws;
  float* wtpad = ws + (size_t)((N + 255) & ~255);
  float* bufA  = wtpad + 6656;
  float* bufB  = bufA + (size_t)N * 80;

  auto cdiv = [](long long a, long long b) { return (int)((a + b - 1) / b); };
  const int T = 256;
  const int gGemm = cdiv(N, 64);    // 64 rows per 128-thread GEMM block

  // degree + dinv
  k_zero<<<cdiv(N, T), T, 0, stream>>>(dinv, N);
  k_degree<<<cdiv(E, T), T, 0, stream>>>(dst, dinv, E);
  k_rsqrt_deg<<<cdiv(N, T), T, 0, stream>>>(dinv, N);

  // layer 1: 64 -> 80   (Kpad=64, Fpad=80)
  k_pack_wt<<<cdiv(64 * 80, T), T, 0, stream>>>(W1, wtpad, 64, 80, 64, 80);
  k_gemm_wmma<<<gGemm, 128, 0, stream>>>(x, wtpad, bufA, N, 64, 80);
  k_zero<<<cdiv((long long)N * 80, T), T, 0, stream>>>(bufB, (long long)N * 80);
  k_aggregate<<<cdiv((long long)E * 20, T), T, 0, stream>>>(bufA, src, dst, dinv, bufB, E, 80, 20);
  k_selfloop_bias_act<<<cdiv((long long)N * 80, T), T, 0, stream>>>(bufA, dinv, b1, bufB, N, 80, 1);

  // layer 2: 80 -> 40   (Kpad=80, Fpad=48)
  k_pack_wt<<<cdiv(80 * 48, T), T, 0, stream>>>(W2, wtpad, 80, 40, 80, 48);
  k_gemm_wmma<<<gGemm, 128, 0, stream>>>(bufB, wtpad, bufA, N, 80, 40);
  k_zero<<<cdiv((long long)N * 40, T), T, 0, stream>>>(bufB, (long long)N * 40);
  k_aggregate<<<cdiv((long long)E * 10, T), T, 0, stream>>>(bufA, src, dst, dinv, bufB, E, 40, 10);
  k_selfloop_bias_act<<<cdiv((long long)N * 40, T), T, 0, stream>>>(bufA, dinv, b2, bufB, N, 40, 1);

  // layer 3: 40 -> 50   (Kpad=40, Fpad=64)
  k_pack_wt<<<cdiv(40 * 64, T), T, 0, stream>>>(W3, wtpad, 40, 50, 40, 64);
  k_gemm_wmma<<<gGemm, 128, 0, stream>>>(bufB, wtpad, bufA, N, 40, 50);
  k_zero<<<cdiv((long long)N * 50, T), T, 0, stream>>>(bufB, (long long)N * 50);
  k_aggregate<<<cdiv((long long)E * 13, T), T, 0, stream>>>(bufA, src, dst, dinv, bufB, E, 50, 13);
  k_selfloop_bias_act<<<cdiv((long long)N * 50, T), T, 0, stream>>>(bufA, dinv, b3, bufB, N, 50, 1);

  // classifier: 50 -> 10 (Kpad=52, Fpad=16), then fused bias + log_softmax
  k_pack_wt<<<cdiv(52 * 16, T), T, 0, stream>>>(Wl, wtpad, 50, 10, 52, 16);
  k_gemm_wmma<<<gGemm, 128, 0, stream>>>(bufB, wtpad, bufA, N, 50, 10);
  k_logsoftmax10<<<cdiv(N, T), T, 0, stream>>>(bufA, bl, (float*)d_out, N);
}